// InclusivityLoss_51462298141282
// MI455X (gfx1250) — compile-verified
//
#include <hip/hip_runtime.h>
#include <hip/hip_bf16.h>

typedef float v2f __attribute__((ext_vector_type(2)));
typedef float v8f __attribute__((ext_vector_type(8)));

#define B_DIM 16
#define M_DIM 128
#define N_DIM 16384

// ---------------------------------------------------------------------------
// Kernel 1: per-point squared norms  norms[b*N + n] = x^2 + y^2 + z^2
// ---------------------------------------------------------------------------
__global__ __launch_bounds__(256)
void pc_norm_kernel(const float* __restrict__ pc, float* __restrict__ norms) {
    int idx = blockIdx.x * blockDim.x + threadIdx.x;   // b*N + n
    if (idx >= B_DIM * N_DIM) return;
    int b = idx / N_DIM;
    int n = idx - b * N_DIM;
    const float* base = pc + (size_t)b * 3 * N_DIM;
    float x = base[n];
    float y = base[N_DIM + n];
    float z = base[2 * N_DIM + n];
    norms[idx] = x * x + y * y + z * z;
}

// ---------------------------------------------------------------------------
// Kernel 2: WMMA min-distance.
// grid = (8 mtiles, 16 batches), block = 256 (8 waves of 32).
// Each wave handles 128 contiguous N-tiles of 16 points with
// V_WMMA_F32_16X16X4_F32 computing  D[m][n] = ||pc_n||^2 - 2*kp_m.pc_n,
// followed by a running v_min into 8 accumulator VGPRs.
// A-frag (16x4 f32, 2 VGPRs): lanes 0-15 hold K=0/1, lanes 16-31 hold K=2/3.
// B-frag (4x16 f32, 2 VGPRs): same K split, N = lane%16.
// C/D (16x16 f32, 8 VGPRs): VGPR g, lane l -> M = g + 8*(l>=16), N = l%16.
// ---------------------------------------------------------------------------
__global__ __launch_bounds__(256)
void min_dist_kernel(const float* __restrict__ kp, const float* __restrict__ pc,
                     const float* __restrict__ norms, float* __restrict__ mind) {
    __shared__ float lds[16][128];   // [m within tile][wave*16 + lane%16]

    const int b     = blockIdx.y;
    const int mtile = blockIdx.x;
    const int tid   = threadIdx.x;
    const int lane  = tid & 31;
    const int wave  = tid >> 5;
    const int l15   = lane & 15;
    const bool hi   = (lane >= 16);

    const float* kpb = kp + (size_t)b * 3 * M_DIM;
    const float* pcb = pc + (size_t)b * 3 * N_DIM;
    const float* nmb = norms + (size_t)b * N_DIM;

    // ---- A fragment: row m = [-2kx, -2ky, -2kz, 1] ----
    const int mg = mtile * 16 + l15;
    const float kx = kpb[mg];
    const float ky = kpb[M_DIM + mg];
    const float kz = kpb[2 * M_DIM + mg];
    v2f a;
    a.x = hi ? (-2.0f * kz) : (-2.0f * kx);   // K = 2 : 0
    a.y = hi ?   1.0f       : (-2.0f * ky);   // K = 3 : 1

    // ---- per-lane B source pointers (select once; uniform loop below) ----
    const float* p0 = hi ? (pcb + 2 * N_DIM) : pcb;           // pcz : pcx
    const float* p1 = hi ? nmb               : (pcb + N_DIM); // norm : pcy

    const float INF = __builtin_inff();
    v8f minA, minB;
    #pragma unroll
    for (int g = 0; g < 8; ++g) { minA[g] = INF; minB[g] = INF; }
    v8f czero = {};

    // wave owns tiles [wave*128, wave*128+128); element index per tile step 16
    const int nbase = wave * 128 * 16 + l15;

    for (int t = 0; t < 128; t += 2) {
        const int n0 = nbase + t * 16;
        const int n1 = n0 + 16;
        v2f b0, b1;
        b0.x = p0[n0];  b0.y = p1[n0];
        b1.x = p0[n1];  b1.y = p1[n1];

        v8f d0 = __builtin_amdgcn_wmma_f32_16x16x4_f32(
                     false, a, false, b0, (short)0, czero, false, false);
        v8f d1 = __builtin_amdgcn_wmma_f32_16x16x4_f32(
                     false, a, false, b1, (short)0, czero, false, false);

        #pragma unroll
        for (int g = 0; g < 8; ++g) {
            minA[g] = fminf(minA[g], d0[g]);
            minB[g] = fminf(minB[g], d1[g]);
        }
    }
    #pragma unroll
    for (int g = 0; g < 8; ++g) minA[g] = fminf(minA[g], minB[g]);

    // ---- cross-lane / cross-wave reduction via LDS ----
    #pragma unroll
    for (int g = 0; g < 8; ++g)
        lds[g + (hi ? 8 : 0)][wave * 16 + l15] = minA[g];
    __syncthreads();

    if (tid < 16) {
        float mn = INF;
        for (int j = 0; j < 128; ++j) mn = fminf(mn, lds[tid][j]);
        const int m_g = mtile * 16 + tid;
        const float x = kpb[m_g];
        const float y = kpb[M_DIM + m_g];
        const float z = kpb[2 * M_DIM + m_g];
        float d2 = mn + (x * x + y * y + z * z);
        d2 = fmaxf(d2, 0.0f);                  // guard tiny negative from cancellation
        mind[b * M_DIM + m_g] = sqrtf(d2);
    }
}

// ---------------------------------------------------------------------------
// Kernel 3: deterministic mean of the 2048 per-(b,m) min distances.
// ---------------------------------------------------------------------------
__global__ __launch_bounds__(256)
void finalize_kernel(const float* __restrict__ mind, float* __restrict__ out) {
    __shared__ float s[256];
    const int tid = threadIdx.x;
    float acc = 0.0f;
    #pragma unroll
    for (int i = 0; i < 8; ++i) acc += mind[tid * 8 + i];   // fixed order
    s[tid] = acc;
    __syncthreads();
    for (int stride = 128; stride > 0; stride >>= 1) {
        if (tid < stride) s[tid] += s[tid + stride];
        __syncthreads();
    }
    if (tid == 0) out[0] = s[0] / (float)(B_DIM * M_DIM);
}

// ---------------------------------------------------------------------------
extern "C" void kernel_launch(void* const* d_in, const int* in_sizes, int n_in,
                              void* d_out, int out_size, void* d_ws, size_t ws_size,
                              hipStream_t stream) {
    const float* kp = (const float*)d_in[0];   // B x 3 x M
    const float* pc = (const float*)d_in[1];   // B x 3 x N
    float* out   = (float*)d_out;              // scalar f32
    float* norms = (float*)d_ws;                       // B*N floats
    float* mind  = norms + (size_t)B_DIM * N_DIM;      // B*M floats

    pc_norm_kernel<<<(B_DIM * N_DIM + 255) / 256, 256, 0, stream>>>(pc, norms);

    dim3 grid(M_DIM / 16, B_DIM);   // (8, 16)
    min_dist_kernel<<<grid, 256, 0, stream>>>(kp, pc, norms, mind);

    finalize_kernel<<<1, 256, 0, stream>>>(mind, out);
}